// GCN_twin_26388279066915
// MI455X (gfx1250) — compile-verified
//
#include <hip/hip_runtime.h>
#include <hip/hip_bf16.h>

typedef float v2f __attribute__((ext_vector_type(2)));
typedef float v8f __attribute__((ext_vector_type(8)));

#define F 128
#define NUM_GRAPHS 64

// ---------------- utility kernels ----------------

__global__ void k_zero(float* __restrict__ p, size_t n) {
    size_t i = (size_t)blockIdx.x * blockDim.x + threadIdx.x;
    size_t st = (size_t)gridDim.x * blockDim.x;
    for (; i < n; i += st) p[i] = 0.0f;
}

__global__ void k_fill(float* __restrict__ p, int n, float v) {
    int i = blockIdx.x * blockDim.x + threadIdx.x;
    if (i < n) p[i] = v;
}

// deg[dst[e]] += 1  (deg pre-filled with 1.0 for the self-loop)
__global__ void k_deg(const int* __restrict__ dst, float* __restrict__ deg, int E) {
    int i = blockIdx.x * blockDim.x + threadIdx.x;
    if (i < E) atomicAdd(&deg[dst[i]], 1.0f);
}

// deg -> deg^-1/2 in place
__global__ void k_rsqrt(float* __restrict__ d, int N) {
    int i = blockIdx.x * blockDim.x + threadIdx.x;
    if (i < N) d[i] = rsqrtf(d[i]);
}

// Wt[n][k] = W[k][n], 128x128
__global__ void k_transpose128(const float* __restrict__ W, float* __restrict__ Wt) {
    int i = blockIdx.x * blockDim.x + threadIdx.x;  // 0..16383
    int k = i >> 7, n = i & 127;
    Wt[n * F + k] = W[i];
}

// ---------------- WMMA GEMM: H[N,128] = X[N,128] @ W[128,128] ----------------
// One block = 16-row M tile, 8 waves each own a 16-col N tile.
// f32 WMMA 16x16x4; A frag: lane l(+16*hi) holds X[row][k0+2*hi + {0,1}]
// B frag (via Wt): Wt[col][k0+2*hi + {0,1}]; D: VGPR r -> row r+8*hi, col l.
__global__ void __launch_bounds__(256) k_gemm_wmma(
    const float* __restrict__ X, const float* __restrict__ Wt,
    float* __restrict__ H, int N)
{
    int lane = threadIdx.x & 31;
    int wave = threadIdx.x >> 5;     // 0..7 -> column tile
    int l  = lane & 15;
    int hi = lane >> 4;              // 0 or 1
    int row0 = blockIdx.x * 16;
    int arow = row0 + l;
    if (arow > N - 1) arow = N - 1;  // clamp (garbage rows masked on store)
    int col = wave * 16 + l;

    const float* xr = X  + (size_t)arow * F;
    const float* wc = Wt + (size_t)col * F;

    v8f acc = {};
    #pragma unroll
    for (int k0 = 0; k0 < F; k0 += 4) {
        int kk = k0 + 2 * hi;
        v2f a, b;
        a.x = xr[kk + 0]; a.y = xr[kk + 1];
        b.x = wc[kk + 0]; b.y = wc[kk + 1];
        acc = __builtin_amdgcn_wmma_f32_16x16x4_f32(
            /*neg_a=*/false, a, /*neg_b=*/false, b,
            /*c_mod=*/(short)0, acc, /*reuse_a=*/false, /*reuse_b=*/false);
    }

    #pragma unroll
    for (int r = 0; r < 8; ++r) {
        int orow = row0 + r + 8 * hi;
        if (orow < N) H[(size_t)orow * F + col] = acc[r];
    }
}

// ---------------- edge scatter: agg[dst] += h[src] * dinv[src]*dinv[dst] ----
// one wave per edge; each lane moves a contiguous float4 (coalesced gather).
__global__ void __launch_bounds__(256) k_scatter(
    const int* __restrict__ src, const int* __restrict__ dst,
    const float* __restrict__ dinv, const float* __restrict__ H,
    float* __restrict__ agg, int E)
{
    int t = blockIdx.x * blockDim.x + threadIdx.x;
    int e = t >> 5, lane = t & 31;
    if (e >= E) return;
    int s = src[e], d = dst[e];
    float norm = dinv[s] * dinv[d];
    const float* hr = H + (size_t)s * F;
    float* ar = agg + (size_t)d * F;
    int f = lane * 4;
    float4 v = *(const float4*)(hr + f);
    atomicAdd(ar + f + 0, v.x * norm);
    atomicAdd(ar + f + 1, v.y * norm);
    atomicAdd(ar + f + 2, v.z * norm);
    atomicAdd(ar + f + 3, v.w * norm);
}

// out = agg + h * dinv^2 + b  (optional relu)
__global__ void k_combine(const float* __restrict__ agg, const float* __restrict__ H,
                          const float* __restrict__ dinv, const float* __restrict__ bias,
                          float* __restrict__ out, int N, int relu)
{
    size_t i = (size_t)blockIdx.x * blockDim.x + threadIdx.x;
    if (i >= (size_t)N * F) return;
    int n = (int)(i >> 7), f = (int)(i & 127);
    float di = dinv[n];
    float v = agg[i] + H[i] * di * di + bias[f];
    if (relu && v < 0.0f) v = 0.0f;
    out[i] = v;
}

// pooled[g][branch*128 + f] += x[n][f]
__global__ void k_pool(const float* __restrict__ x, const int* __restrict__ batch,
                       float* __restrict__ pooled, int branch, int N)
{
    size_t i = (size_t)blockIdx.x * blockDim.x + threadIdx.x;
    if (i >= (size_t)N * F) return;
    int n = (int)(i >> 7), f = (int)(i & 127);
    int g = batch[n];
    atomicAdd(&pooled[(size_t)g * (2 * F) + branch * F + f], x[i]);
}

__global__ void k_count(const int* __restrict__ batch, float* __restrict__ cnt, int N) {
    int i = blockIdx.x * blockDim.x + threadIdx.x;
    if (i < N) atomicAdd(&cnt[batch[i]], 1.0f);
}

// out[g][j] = sum_k (pooled[g][k]/cnt) * WL[k][j] + bL[j]   (64x2 outputs)
__global__ void k_final(const float* __restrict__ pooled, const float* __restrict__ cnt,
                        const float* __restrict__ WL, const float* __restrict__ bL,
                        float* __restrict__ out)
{
    int t = threadIdx.x;            // 0..127
    if (t >= NUM_GRAPHS * 2) return;
    int g = t >> 1, j = t & 1;
    float c0 = fmaxf(cnt[g], 1.0f);
    float c1 = fmaxf(cnt[NUM_GRAPHS + g], 1.0f);
    float s = 0.0f;
    for (int k = 0; k < 2 * F; ++k) {
        float feat = pooled[g * (2 * F) + k] / (k < F ? c0 : c1);
        s += feat * WL[k * 2 + j];
    }
    out[g * 2 + j] = s + bL[j];
}

// ---------------- host side ----------------

extern "C" void kernel_launch(void* const* d_in, const int* in_sizes, int n_in,
                              void* d_out, int out_size, void* d_ws, size_t ws_size,
                              hipStream_t stream)
{
    const float* x[2]     = { (const float*)d_in[0], (const float*)d_in[3] };
    const int*   ei[2]    = { (const int*)d_in[2],   (const int*)d_in[5]   };
    const int*   batch[2] = { (const int*)d_in[6],   (const int*)d_in[7]   };
    const float* Wl[3] = { (const float*)d_in[8], (const float*)d_in[10], (const float*)d_in[12] };
    const float* bl[3] = { (const float*)d_in[9], (const float*)d_in[11], (const float*)d_in[13] };
    const float* WL = (const float*)d_in[14];
    const float* bL = (const float*)d_in[15];
    float* out = (float*)d_out;

    const int N = in_sizes[0] / F;      // 50000
    const int E = in_sizes[2] / 2;      // 500000
    const size_t NF = (size_t)N * F;

    // workspace layout
    float* ws     = (float*)d_ws;
    float* hbuf   = ws;                 // N*F
    float* xbuf   = hbuf + NF;          // N*F
    float* aggbuf = xbuf + NF;          // N*F
    float* Wt     = aggbuf + NF;        // 128*128
    float* deg    = Wt + F * F;         // N (becomes dinv)
    float* pooled = deg + N;            // 64*256
    float* cnt    = pooled + NUM_GRAPHS * 2 * F;  // 128 (64 per branch)

    const int T = 256;
    const int gNF  = (int)((NF + T - 1) / T);
    const int gN   = (N + T - 1) / T;
    const int gE   = (E + T - 1) / T;
    const int gE32 = (int)(((size_t)E * 32 + T - 1) / T);
    const int gM   = (N + 15) / 16;

    // zero pooled sums + counts
    k_zero<<<(NUM_GRAPHS * 2 * F + 128 + T - 1) / T, T, 0, stream>>>(
        pooled, (size_t)(NUM_GRAPHS * 2 * F + 128));

    for (int b = 0; b < 2; ++b) {
        const int* srcI = ei[b];
        const int* dstI = ei[b] + E;

        // degrees with self loop, then deg^-1/2
        k_fill<<<gN, T, 0, stream>>>(deg, N, 1.0f);
        k_deg<<<gE, T, 0, stream>>>(dstI, deg, E);
        k_rsqrt<<<gN, T, 0, stream>>>(deg, N);

        const float* cur = x[b];
        for (int l = 0; l < 3; ++l) {
            k_transpose128<<<(F * F + T - 1) / T, T, 0, stream>>>(Wl[l], Wt);
            k_gemm_wmma<<<gM, T, 0, stream>>>(cur, Wt, hbuf, N);
            k_zero<<<gNF, T, 0, stream>>>(aggbuf, NF);
            k_scatter<<<gE32, T, 0, stream>>>(srcI, dstI, deg, hbuf, aggbuf, E);
            k_combine<<<gNF, T, 0, stream>>>(aggbuf, hbuf, deg, bl[l], xbuf, N, l < 2 ? 1 : 0);
            cur = xbuf;
        }

        k_count<<<gN, T, 0, stream>>>(batch[b], cnt + b * NUM_GRAPHS, N);
        k_pool<<<gNF, T, 0, stream>>>(xbuf, batch[b], pooled, b, N);
    }

    k_final<<<1, 128, 0, stream>>>(pooled, cnt, WL, bL, out);
}